// MBoundaryLoss_74165495267538
// MI455X (gfx1250) — compile-verified
//
#include <hip/hip_runtime.h>
#include <math.h>

typedef float v2f __attribute__((ext_vector_type(2)));
typedef float v8f __attribute__((ext_vector_type(8)));
typedef int vint4 __attribute__((__vector_size__(16)));

#define B_N 8192
#define C_N 10
#define D_N 256
#define EPS_JITTER 1e-11f
#define QB_WAVES 4               // waves (b-tiles) per block in quad kernel
#define STRIP_PITCH 272          // 256 + 16 pad: half-waves hit disjoint LDS banks

#if defined(__has_builtin)
#if __has_builtin(__builtin_amdgcn_global_load_async_to_lds_b128) && \
    __has_builtin(__builtin_amdgcn_s_wait_asynccnt)
#define USE_ASYNC_LDS 1
#endif
#endif

__device__ __forceinline__ float softplus_f(float x) {
    if (x > 20.f) return x;
    if (x < -20.f) return expf(x);
    return log1pf(expf(x));
}

// out[1..11) = softplus(delta); out[0] finalized by loss kernels.
__global__ void init_out_kernel(const float* __restrict__ delta, float* __restrict__ out) {
    int t = threadIdx.x;
    if (t < C_N) out[1 + t] = softplus_f(delta[t]);
}

// Deterministic per-class index compaction (preserves row order -> stable f32 sums).
__global__ void compact_kernel(const int* __restrict__ labels,
                               int* __restrict__ idx, int* __restrict__ cls_cnt) {
    int c = blockIdx.x;
    if (threadIdx.x == 0) {
        int* my = idx + (size_t)c * B_N;
        int cnt = 0;
        for (int n = 0; n < B_N; ++n)
            if (labels[n] == c) my[cnt++] = n;
        cls_cnt[c] = cnt;
    }
}

// One block per class; thread d accumulates dim d of the class mean over gathered rows.
__global__ void means_kernel(const float* __restrict__ src, const int* __restrict__ idx,
                             const int* __restrict__ cls_cnt, float* __restrict__ mu) {
    int c = blockIdx.x, d = threadIdx.x;
    const int* my = idx + (size_t)c * B_N;
    int cnt = cls_cnt[c];
    float s = 0.f;
    for (int i = 0; i < cnt; ++i) s += src[(size_t)my[i] * D_N + d];
    mu[c * D_N + d] = s / (float)cnt;
}

// cov_c = sum over gathered class rows of centered outer-products / (cnt-1).
// 16x16 tile per wave, K accumulated 4 rows at a time via v_wmma_f32_16x16x4_f32.
// A-layout (16x4 f32): lane l holds M=l%16, K=(l/16)*2+{0,1}; B analogous with N=l%16.
__global__ void cov_wmma_kernel(const float* __restrict__ src, const int* __restrict__ idx,
                                const int* __restrict__ cls_cnt, const float* __restrict__ mu,
                                float* __restrict__ cov) {
    int c  = blockIdx.z;
    int d0 = blockIdx.x * 16, e0 = blockIdx.y * 16;
    int lane = threadIdx.x;
    int m = lane & 15, kh = lane >> 4;
    const int* my = idx + (size_t)c * B_N;
    int cnt = cls_cnt[c];
    const float* muc = mu + c * D_N;
    float mud = muc[d0 + m], mue = muc[e0 + m];
    v8f acc = {};
    for (int t = 0; t < cnt; t += 4) {
        int i0 = t + kh * 2, i1 = i0 + 1;
        float mk0 = (i0 < cnt) ? 1.f : 0.f;
        float mk1 = (i1 < cnt) ? 1.f : 0.f;
        int r0 = (i0 < cnt) ? my[i0] : 0;
        int r1 = (i1 < cnt) ? my[i1] : 0;
        v2f a, bb;
        a.x  = mk0 * (src[(size_t)r0 * D_N + d0 + m] - mud);
        a.y  = mk1 * (src[(size_t)r1 * D_N + d0 + m] - mud);
        bb.x = mk0 * (src[(size_t)r0 * D_N + e0 + m] - mue);
        bb.y = mk1 * (src[(size_t)r1 * D_N + e0 + m] - mue);
        acc = __builtin_amdgcn_wmma_f32_16x16x4_f32(false, a, false, bb,
                                                    (short)0, acc, false, false);
    }
    float norm = 1.0f / ((float)cnt - 1.0f);
    #pragma unroll
    for (int v = 0; v < 8; ++v) {
        int row = d0 + v + kh * 8;
        cov[(size_t)c * D_N * D_N + (size_t)row * D_N + e0 + m] = acc[v] * norm;
    }
}

// In-place Gauss-Jordan inversion of (cov + eps*I): full 256x256 f32 matrix in LDS.
// 256x257 padded + 256 pivot column = 264,192 B -- fits only in CDNA5's 320 KB WGP LDS.
__global__ void gj_invert_kernel(float* __restrict__ cov) {
    extern __shared__ float lds[];
    float* colk = lds + 256 * 257;
    int c = blockIdx.x, j = threadIdx.x;
    float* A = cov + (size_t)c * D_N * D_N;
    for (int i = 0; i < D_N; ++i)
        lds[i * 257 + j] = A[(size_t)i * D_N + j] + ((i == j) ? EPS_JITTER : 0.f);
    __syncthreads();
    for (int k = 0; k < D_N; ++k) {
        colk[j] = lds[j * 257 + k];         // thread index as row here
        float rowkj = lds[k * 257 + j];     // pivot-row element in my column
        __syncthreads();
        float pivinv = 1.0f / colk[k];
        float nr = ((j == k) ? 1.0f : rowkj) * pivinv;
        for (int i = 0; i < D_N; ++i) {
            if (i == k) {
                lds[i * 257 + j] = nr;
            } else {
                float old = (j == k) ? 0.f : lds[i * 257 + j];
                lds[i * 257 + j] = old - colk[i] * nr;
            }
        }
        __syncthreads();
    }
    for (int i = 0; i < D_N; ++i)
        A[(size_t)i * D_N + j] = lds[i * 257 + j];
}

// Stage a 4x256 f32 strip of Minv into padded LDS (1024 floats, 128 threads).
#if defined(USE_ASYNC_LDS)
__device__ __forceinline__ void stage_strip_async(float* buf, const float* gsrc, int tid) {
    int e0 = tid * 4;
    int r0 = e0 >> 8, c0 = e0 & 255;
    __builtin_amdgcn_global_load_async_to_lds_b128(
        (__attribute__((address_space(1))) vint4*)(gsrc + e0),
        (__attribute__((address_space(3))) vint4*)(buf + r0 * STRIP_PITCH + c0), 0, 0);
    int e1 = e0 + 512;
    int r1 = e1 >> 8, c1 = e1 & 255;
    __builtin_amdgcn_global_load_async_to_lds_b128(
        (__attribute__((address_space(1))) vint4*)(gsrc + e1),
        (__attribute__((address_space(3))) vint4*)(buf + r1 * STRIP_PITCH + c1), 0, 0);
}
#else
__device__ __forceinline__ void stage_strip_sync(float* buf, const float* gsrc, int tid) {
    int e0 = tid * 4;
    int r0 = e0 >> 8, c0 = e0 & 255;
    *(float4*)(buf + r0 * STRIP_PITCH + c0) = *(const float4*)(gsrc + e0);
    int e1 = e0 + 512;
    int r1 = e1 >> 8, c1 = e1 & 255;
    *(float4*)(buf + r1 * STRIP_PITCH + c1) = *(const float4*)(gsrc + e1);
}
#endif

// dis[b,c] = z^T Minv z. Block = 4 waves sharing one class; each wave owns a 16-row
// b-tile and 16 v8f accumulators (full 16x256 Y strip). Per 4-wide k-strip the block
// stages Minv[k..k+4) into LDS -- double-buffered via async-to-LDS DMA when available
// (ASYNCcnt + one barrier per step), else synchronous b128 copies + prefetch.
__global__ void quad_wmma_kernel(const float* __restrict__ pooled,
                                 const float* __restrict__ centroids,
                                 const float* __restrict__ invcov,
                                 float* __restrict__ out) {
    __shared__ float strip[2][4 * STRIP_PITCH];
    int c = blockIdx.y;
    int tid = threadIdx.x;            // 0..127
    int wave = tid >> 5, lane = tid & 31;
    int m = lane & 15, kh = lane >> 4;
    int b0 = blockIdx.x * (16 * QB_WAVES) + wave * 16;
    const float* Mi   = invcov + (size_t)c * D_N * D_N;
    const float* cen  = centroids + c * D_N;
    const float* prow = pooled + (size_t)(b0 + m) * D_N;

    v8f acc[16];
    #pragma unroll
    for (int nt = 0; nt < 16; ++nt) acc[nt] = (v8f){};

#if defined(USE_ASYNC_LDS)
    stage_strip_async(strip[0], Mi, tid);     // prime buffer 0
#endif
    for (int k = 0; k < D_N; k += 4) {
        int cur = (k >> 2) & 1;
#if defined(USE_ASYNC_LDS)
        __builtin_amdgcn_s_wait_asynccnt(0);  // my DMA done
        __syncthreads();                      // whole block: strip[cur] ready, prev reads done
        if (k + 4 < D_N)                      // overlap next strip's DMA with this WMMA chain
            stage_strip_async(strip[cur ^ 1], Mi + (size_t)(k + 4) * D_N, tid);
#else
        __syncthreads();                      // prev iteration's strip reads complete
        stage_strip_sync(strip[cur], Mi + (size_t)k * D_N, tid);
        if (k + 4 < D_N)                      // pull next strip toward caches early
            __builtin_prefetch(Mi + (size_t)(k + 4) * D_N + tid * 8, 0, 1);
        __syncthreads();
#endif
        int kk = k + kh * 2;
        v2f a;
        a.x = prow[kk]     - cen[kk];
        a.y = prow[kk + 1] - cen[kk + 1];
        const float* srow0 = strip[cur] + (kh * 2) * STRIP_PITCH;
        const float* srow1 = srow0 + STRIP_PITCH;
        #pragma unroll
        for (int nt = 0; nt < 16; ++nt) {
            v2f bb;
            bb.x = srow0[nt * 16 + m];
            bb.y = srow1[nt * 16 + m];
            acc[nt] = __builtin_amdgcn_wmma_f32_16x16x4_f32(false, a, false, bb,
                                                            (short)0, acc[nt], false, false);
        }
    }

    // part[v] = sum_n Y[m_v, n] * z[m_v, n]
    float part[8];
    #pragma unroll
    for (int v = 0; v < 8; ++v) part[v] = 0.f;
    #pragma unroll
    for (int nt = 0; nt < 16; ++nt) {
        int col = nt * 16 + m;
        float zc = cen[col];
        #pragma unroll
        for (int v = 0; v < 8; ++v) {
            int row = b0 + v + kh * 8;
            float z = pooled[(size_t)row * D_N + col] - zc;
            part[v] += acc[nt][v] * z;
        }
    }
    // butterfly reduce over the 16 lanes within each half-wave
    #pragma unroll
    for (int off = 8; off > 0; off >>= 1) {
        #pragma unroll
        for (int v = 0; v < 8; ++v)
            part[v] += __shfl_xor(part[v], off, 16);
    }
    if (m == 0) {
        float* logits = out + 1 + C_N;   // [B, C] row-major
        #pragma unroll
        for (int v = 0; v < 8; ++v) {
            int row = b0 + v + kh * 8;
            logits[(size_t)row * C_N + c] = truncf(part[v]);
        }
    }
}

// Per-block partial sums of ((euc-d)*pos + (d-euc)*neg); deterministic (no atomics).
__global__ void loss_kernel(const float* __restrict__ pooled,
                            const float* __restrict__ centroids,
                            const int* __restrict__ labels,
                            const float* __restrict__ delta,
                            float* __restrict__ partial) {
    __shared__ float red[256];
    int b = blockIdx.x * 256 + threadIdx.x;
    int lb = labels[b];
    const float* cen = centroids + lb * D_N;
    const float* p   = pooled + (size_t)b * D_N;
    float ss = 0.f;
    for (int d = 0; d < D_N; ++d) { float df = p[d] - cen[d]; ss += df * df; }
    float euc = sqrtf(ss);
    float dsp = softplus_f(delta[lb]);
    float contrib = 0.f;
    if (euc > dsp)      contrib = euc - dsp;
    else if (euc < dsp) contrib = dsp - euc;
    red[threadIdx.x] = contrib;
    __syncthreads();
    for (int off = 128; off > 0; off >>= 1) {
        if (threadIdx.x < off) red[threadIdx.x] += red[threadIdx.x + off];
        __syncthreads();
    }
    if (threadIdx.x == 0) partial[blockIdx.x] = red[0];
}

__global__ void loss_finalize_kernel(const float* __restrict__ partial, float* __restrict__ out) {
    if (threadIdx.x == 0) {
        float s = 0.f;
        for (int i = 0; i < B_N / 256; ++i) s += partial[i];
        out[0] = s * (1.0f / (float)B_N);
    }
}

extern "C" void kernel_launch(void* const* d_in, const int* in_sizes, int n_in,
                              void* d_out, int out_size, void* d_ws, size_t ws_size,
                              hipStream_t stream) {
    const float* pooled    = (const float*)d_in[0];
    const float* centroids = (const float*)d_in[1];
    const int*   labels    = (const int*)d_in[2];
    const float* src       = (const float*)d_in[3];
    const float* delta     = (const float*)d_in[4];
    float* out = (float*)d_out;
    float* ws  = (float*)d_ws;

    float* mu       = ws;                                   // 2560 f
    int*   cls_cnt  = (int*)(ws + 2560);                    // 16 ints (10 used)
    int*   idx      = (int*)(ws + 2576);                    // 10*8192 ints
    float* cov      = ws + 2576 + (size_t)C_N * B_N;        // 10*256*256 f (inverted in place)
    float* losspart = cov + (size_t)C_N * D_N * D_N;        // 32 partials
    (void)in_sizes; (void)n_in; (void)out_size; (void)ws_size;

    init_out_kernel<<<1, 32, 0, stream>>>(delta, out);
    compact_kernel<<<C_N, 32, 0, stream>>>(labels, idx, cls_cnt);
    means_kernel<<<C_N, 256, 0, stream>>>(src, idx, cls_cnt, mu);
    cov_wmma_kernel<<<dim3(16, 16, C_N), 32, 0, stream>>>(src, idx, cls_cnt, mu, cov);
    size_t ldsBytes = (size_t)(256 * 257 + 256) * sizeof(float); // 264,192 B <= 320 KB
    gj_invert_kernel<<<C_N, 256, ldsBytes, stream>>>(cov);
    quad_wmma_kernel<<<dim3(B_N / (16 * QB_WAVES), C_N), 32 * QB_WAVES, 0, stream>>>(
        pooled, centroids, cov, out);
    loss_kernel<<<B_N / 256, 256, 0, stream>>>(pooled, centroids, labels, delta, losspart);
    loss_finalize_kernel<<<1, 32, 0, stream>>>(losspart, out);
}